// Network_86517821214889
// MI455X (gfx1250) — compile-verified
//
#include <hip/hip_runtime.h>
#include <hip/hip_bf16.h>
#include <stdint.h>

// ---------------------------------------------------------------------------
// RPN anchor-target op for MI455X (gfx1250).
// Store-bandwidth bound: ~95 MB of output (floor ~4.1 us @ 23.3 TB/s).
// No matmul structure -> WMMA not applicable; CDNA5 paths used instead:
//   * global_load_async_to_lds_b128 + s_wait_asynccnt (ASYNCcnt async copy)
//   * wave32 shfl butterfly argmax (ds_swizzle/permlane), 1 barrier/16 rows
// ---------------------------------------------------------------------------

namespace {
constexpr int   kStride     = 16;
constexpr int   kFeatW      = 100;
constexpr int   kFeatH      = 100;
constexpr int   kNumA       = 9;
constexpr int   kNumAnchors = kFeatH * kFeatW * kNumA;   // 90000
constexpr int   kNumGT      = 256;
constexpr int   kRows       = 16;                        // anchor rows / block
constexpr int   kBlocks     = kNumAnchors / kRows;       // 5625
constexpr float kImgWm1     = 1600.0f - 1.0f;
constexpr float kImgHm1     = 1600.0f - 1.0f;
}

// Base anchors for 16x16 base, ratios {0.5,1,2} x scales {8,16,32}
// (exact values produced by _generate_anchors(); all integers).
__device__ __constant__ float c_base[kNumA][4] = {
    { -84.f,  -40.f,  99.f,  55.f},   // r=0.5, s=8
    {-176.f,  -88.f, 191.f, 103.f},   // r=0.5, s=16
    {-360.f, -184.f, 375.f, 199.f},   // r=0.5, s=32
    { -56.f,  -56.f,  71.f,  71.f},   // r=1.0, s=8
    {-120.f, -120.f, 135.f, 135.f},   // r=1.0, s=16
    {-248.f, -248.f, 263.f, 263.f},   // r=1.0, s=32
    { -36.f,  -80.f,  51.f,  95.f},   // r=2.0, s=8
    { -80.f, -168.f,  95.f, 183.f},   // r=2.0, s=16
    {-168.f, -344.f, 183.f, 359.f}    // r=2.0, s=32
};

__global__ __launch_bounds__(kNumGT)
void rpn_target_kernel(const float* __restrict__ deltas,
                       const float* __restrict__ gt_boxes,
                       float* __restrict__ out_prop,   // [90000*4]
                       float* __restrict__ out_ovl,    // [90000*256]
                       float* __restrict__ out_gtd)    // [90000*4]
{
    __shared__ float4 s_gt[kNumGT];          // 4 KB: whole gt_boxes table
    __shared__ float  s_pval[kRows][8];      // per-row, per-wave partial max
    __shared__ int    s_pidx[kRows][8];      // per-row, per-wave partial argmax

    const int tid  = threadIdx.x;
    const int lane = tid & 31;
    const int wave = tid >> 5;
    const int base = blockIdx.x * kRows;

    // ---- CDNA5 async copy: gt_boxes (4 KB) -> LDS once per block,
    //      16 B per lane, tracked by ASYNCcnt. ----
    {
        unsigned lds_addr  = (unsigned)(uintptr_t)(&s_gt[tid]);
        uint64_t glob_addr = (uint64_t)(uintptr_t)(gt_boxes + (size_t)tid * 4);
        asm volatile("global_load_async_to_lds_b128 %0, %1, off"
                     :: "v"(lds_addr), "v"(glob_addr)
                     : "memory");
        asm volatile("s_wait_asynccnt 0" ::: "memory");
    }
    __syncthreads();

    // Lane-owned GT box (fixed across all rows) and its area.
    const float4 g     = s_gt[tid];
    const float  area2 = (g.z - g.x + 1.0f) * (g.w - g.y + 1.0f);

    // Saved state for the row this thread will finalize (row == tid < 16).
    float4 my_prop = make_float4(0.f, 0.f, 0.f, 0.f);
    float  my_aw = 1.f, my_ah = 1.f, my_acx = 0.f, my_acy = 0.f;

    for (int r = 0; r < kRows; ++r) {
        const int n = base + r;

        // ---- Anchor n (regenerated analytically; exact) ----
        const int   a    = n % kNumA;
        const int   site = n / kNumA;
        const float sx   = (float)((site % kFeatW) * kStride);
        const float sy   = (float)((site / kFeatW) * kStride);
        const float ax1  = c_base[a][0] + sx;
        const float ay1  = c_base[a][1] + sy;
        const float ax2  = c_base[a][2] + sx;
        const float ay2  = c_base[a][3] + sy;

        const float aw  = ax2 - ax1 + 1.0f;
        const float ah  = ay2 - ay1 + 1.0f;
        const float acx = ax1 + 0.5f * aw;
        const float acy = ay1 + 0.5f * ah;

        // ---- bbox_transform_inv + clip (uniform across block; s_load'd) ----
        const float4 d  = reinterpret_cast<const float4*>(deltas)[n];
        const float pcx = d.x * aw + acx;
        const float pcy = d.y * ah + acy;
        const float pw  = expf(d.z) * aw;
        const float ph  = expf(d.w) * ah;

        float px1 = pcx - 0.5f * pw;
        float py1 = pcy - 0.5f * ph;
        float px2 = pcx + 0.5f * pw - 1.0f;
        float py2 = pcy + 0.5f * ph - 1.0f;
        px1 = fminf(fmaxf(px1, 0.0f), kImgWm1);
        py1 = fminf(fmaxf(py1, 0.0f), kImgHm1);
        px2 = fminf(fmaxf(px2, 0.0f), kImgWm1);
        py2 = fminf(fmaxf(py2, 0.0f), kImgHm1);

        if (tid == r) {   // keep this row's state for the epilogue
            my_prop = make_float4(px1, py1, px2, py2);
            my_aw = aw; my_ah = ah; my_acx = acx; my_acy = acy;
        }

        // ---- IoU: lane tid vs its GT; coalesced row store (128 B / wave) ----
        const float area1 = (px2 - px1 + 1.0f) * (py2 - py1 + 1.0f);
        const float iw = fmaxf(fminf(px2, g.z) - fmaxf(px1, g.x) + 1.0f, 0.0f);
        const float ih = fmaxf(fminf(py2, g.w) - fmaxf(py1, g.y) + 1.0f, 0.0f);
        const float inter = iw * ih;
        const float iou   = inter / (area1 + area2 - inter);
        out_ovl[(size_t)n * kNumGT + tid] = iou;

        // ---- wave32 butterfly argmax (max value, min index on ties):
        //      associative+commutative, preserves jnp.argmax first-occurrence.
        float bv = iou;
        int   bi = tid;
#pragma unroll
        for (int m = 16; m > 0; m >>= 1) {
            const float ov = __shfl_xor(bv, m, 32);
            const int   oi = __shfl_xor(bi, m, 32);
            if (ov > bv || (ov == bv && oi < bi)) { bv = ov; bi = oi; }
        }
        if (lane == 0) { s_pval[r][wave] = bv; s_pidx[r][wave] = bi; }
    }

    __syncthreads();   // single block barrier for all 16 rows' partials

    // ---- Epilogue: thread t finalizes row t (t < 16) ----
    if (tid < kRows) {
        float bv = s_pval[tid][0];
        int   bi = s_pidx[tid][0];
#pragma unroll
        for (int w2 = 1; w2 < 8; ++w2) {
            const float ov = s_pval[tid][w2];
            const int   oi = s_pidx[tid][w2];
            if (ov > bv || (ov == bv && oi < bi)) { bv = ov; bi = oi; }
        }

        const int n = base + tid;
        reinterpret_cast<float4*>(out_prop)[n] = my_prop;

        const float4 bg  = s_gt[bi];
        const float  gw  = bg.z - bg.x + 1.0f;
        const float  gh  = bg.w - bg.y + 1.0f;
        const float  gcx = bg.x + 0.5f * gw;
        const float  gcy = bg.y + 0.5f * gh;
        reinterpret_cast<float4*>(out_gtd)[n] =
            make_float4((gcx - my_acx) / my_aw, (gcy - my_acy) / my_ah,
                        logf(gw / my_aw), logf(gh / my_ah));
    }
}

extern "C" void kernel_launch(void* const* d_in, const int* in_sizes, int n_in,
                              void* d_out, int out_size, void* d_ws, size_t ws_size,
                              hipStream_t stream) {
    (void)in_sizes; (void)n_in; (void)d_ws; (void)ws_size; (void)out_size;

    const float* deltas   = (const float*)d_in[0];   // [90000,4] f32
    const float* gt_boxes = (const float*)d_in[1];   // [256,4]   f32

    float* out      = (float*)d_out;
    float* out_prop = out;                                           // 360000
    float* out_ovl  = out + (size_t)kNumAnchors * 4;                 // 23,040,000
    float* out_gtd  = out + (size_t)kNumAnchors * 4
                          + (size_t)kNumAnchors * kNumGT;            // 360000

    rpn_target_kernel<<<dim3(kBlocks), dim3(kNumGT), 0, stream>>>(
        deltas, gt_boxes, out_prop, out_ovl, out_gtd);
}